// RoiAlign_15169824489483
// MI455X (gfx1250) — compile-verified
//
#include <hip/hip_runtime.h>
#include <cstdint>
#include <cstddef>

typedef float v4f __attribute__((ext_vector_type(4)));

namespace {

constexpr int kB = 4, kN = 256, kH = 256, kW = 256, kC = 256;
constexpr int kCrop = 14;
constexpr int kPos = kCrop * kCrop;  // 196
constexpr int kGrp = kPos / 4;       // 49 groups of 4 positions

struct Samp {
  uint32_t tl, tr, bl, br;  // element offsets into the image
  float wtl, wtr, wbl, wbr; // bilinear weights (zeroed when out of range)
};

__device__ __forceinline__ Samp make_samp(float bx1, float by1, float bx2,
                                          float by2, int p) {
  const int j = p / kCrop;
  const int i = p - j * kCrop;
  const float t = 1.0f / (float)(kCrop - 1);
  // Reference: ys = y1 + (y2 - 1 - y1) * j/13  (after TF box normalization)
  const float ys = by1 + (by2 - 1.0f - by1) * ((float)j * t);
  const float xs = bx1 + (bx2 - 1.0f - bx1) * ((float)i * t);
  const bool valid = (ys >= 0.0f) && (ys <= (float)(kH - 1)) &&
                     (xs >= 0.0f) && (xs <= (float)(kW - 1));
  const float y0f = floorf(ys);
  const float x0f = floorf(xs);
  const float wy = ys - y0f;
  const float wx = xs - x0f;
  const int y0 = min(max((int)y0f, 0), kH - 1);
  const int y1 = min(max((int)y0f + 1, 0), kH - 1);
  const int x0 = min(max((int)x0f, 0), kW - 1);
  const int x1 = min(max((int)x0f + 1, 0), kW - 1);
  Samp s;
  s.tl = (uint32_t)((y0 * kW + x0) * kC);
  s.tr = (uint32_t)((y0 * kW + x1) * kC);
  s.bl = (uint32_t)((y1 * kW + x0) * kC);
  s.br = (uint32_t)((y1 * kW + x1) * kC);
  const float m = valid ? 1.0f : 0.0f;  // extrapolation_value = 0
  s.wtl = m * (1.0f - wy) * (1.0f - wx);
  s.wtr = m * (1.0f - wy) * wx;
  s.wbl = m * wy * (1.0f - wx);
  s.wbr = m * wy * wx;
  return s;
}

// Per-lane async DMA: 16 bytes global -> LDS, tracked by ASYNCcnt.
__device__ __forceinline__ void async16(uint32_t lds_byte_off,
                                        const float* g) {
  asm volatile("global_load_async_to_lds_b128 %0, %1, off"
               :
               : "v"(lds_byte_off), "v"((uint64_t)(uintptr_t)g)
               : "memory");
}

__global__ __launch_bounds__(256) void roi_align_kernel(
    const float* __restrict__ boxes, const float* __restrict__ fpn,
    float* __restrict__ out) {
  // [buf][position-in-group][corner][C] = 32 KB double buffer
  __shared__ float lds[2][4][4][kC];

  const int box = blockIdx.x;  // b*kN + n
  const int b = box >> 8;      // / kN
  const float* img = fpn + (size_t)b * kH * kW * kC;

  const float bx1 = boxes[box * 4 + 0];
  const float by1 = boxes[box * 4 + 1];
  const float bx2 = boxes[box * 4 + 2];
  const float by2 = boxes[box * 4 + 3];

  const int tid = threadIdx.x;
  const int sub = tid >> 6;       // position within group (0..3)
  const int c = (tid & 63) * 4;   // channel quad

  float* outp = out + (size_t)box * kPos * kC + c;

  auto stage = [&](const Samp& s, int buf) {
    const uint32_t l = (uint32_t)(uintptr_t)&lds[buf][sub][0][c];
    constexpr uint32_t rb = kC * sizeof(float);  // 1 KB corner stride
    async16(l + 0 * rb, img + s.tl + c);
    async16(l + 1 * rb, img + s.tr + c);
    async16(l + 2 * rb, img + s.bl + c);
    async16(l + 3 * rb, img + s.br + c);
  };

  Samp cur = make_samp(bx1, by1, bx2, by2, sub);
  stage(cur, 0);

  for (int g = 0; g < kGrp; ++g) {
    Samp nxt = cur;
    if (g + 1 < kGrp) {
      nxt = make_samp(bx1, by1, bx2, by2, 4 * (g + 1) + sub);
      stage(nxt, (g + 1) & 1);  // prefetch next group (4 more in flight)
      asm volatile("s_wait_asynccnt 0x4" ::: "memory");  // group g landed
    } else {
      asm volatile("s_wait_asynccnt 0x0" ::: "memory");
    }

    const int buf = g & 1;
    const v4f tl = *(const v4f*)&lds[buf][sub][0][c];
    const v4f tr = *(const v4f*)&lds[buf][sub][1][c];
    const v4f bl = *(const v4f*)&lds[buf][sub][2][c];
    const v4f br = *(const v4f*)&lds[buf][sub][3][c];

    v4f r = cur.wtl * tl + cur.wtr * tr + cur.wbl * bl + cur.wbr * br;

    // Streaming output: keep the image working set resident in L2.
    __builtin_nontemporal_store(r, (v4f*)(outp + (size_t)(4 * g + sub) * kC));
    cur = nxt;
  }
}

}  // namespace

extern "C" void kernel_launch(void* const* d_in, const int* in_sizes, int n_in,
                              void* d_out, int out_size, void* d_ws,
                              size_t ws_size, hipStream_t stream) {
  (void)in_sizes; (void)n_in; (void)out_size; (void)d_ws; (void)ws_size;
  const float* boxes = (const float*)d_in[0];  // [4,256,4]
  const float* fpn = (const float*)d_in[1];    // [4,256,256,256]
  float* out = (float*)d_out;                  // [4,256,14,14,256]
  roi_align_kernel<<<kB * kN, 256, 0, stream>>>(boxes, fpn, out);
}